// VolumetricSMPL_5583457484873
// MI455X (gfx1250) — compile-verified
//
#include <hip/hip_runtime.h>
#include <math.h>

#define RES   64
#define NVOX  (RES*RES*RES)      // 262144 voxels
#define NVOX4 (NVOX/4)
#define NVERT 6890
#define NJ    24
#define COEFFN 218               // 1 (closest) + 10 (beta) + 207 (pose_map)

typedef __attribute__((ext_vector_type(2))) float v2f;
typedef __attribute__((ext_vector_type(8))) float v8f;

struct alignas(16) F4 { float v[4]; };

__device__ __forceinline__ void mat4mul(float* __restrict__ o,
                                        const float* __restrict__ a,
                                        const float* __restrict__ b) {
#pragma unroll
  for (int r = 0; r < 4; ++r)
#pragma unroll
    for (int c = 0; c < 4; ++c)
      o[r*4+c] = a[r*4+0]*b[0*4+c] + a[r*4+1]*b[1*4+c] +
                 a[r*4+2]*b[2*4+c] + a[r*4+3]*b[3*4+c];
}

// ---------------------------------------------------------------------------
// Kernel 1: per-batch SMPL prep — Rodrigues, pose_map, joint regression,
// kinematic chain. Emits coeff[b][218] = [1 | beta | pose_map] and
// A16[b][16][24] (component-major relative joint transforms).
// ---------------------------------------------------------------------------
__global__ __launch_bounds__(256)
void prep_kernel(const float* __restrict__ pose, const float* __restrict__ beta,
                 const float* __restrict__ v_template,
                 const float* __restrict__ smpl_shapedirs,
                 const float* __restrict__ Jreg,
                 float* __restrict__ coeff, float* __restrict__ A16w)
{
  const int b   = blockIdx.x;
  const int tid = threadIdx.x;
  __shared__ float sR[NJ*9];
  __shared__ float sBeta[10];
  __shared__ float sJ[NJ*3];
  __shared__ float sPart[8][NJ*3];

  if (tid < 10) sBeta[tid] = beta[b*10 + tid];
  if (tid < NJ) {
    const float r0 = pose[b*72 + tid*3 + 0];
    const float r1 = pose[b*72 + tid*3 + 1];
    const float r2 = pose[b*72 + tid*3 + 2];
    const float th  = sqrtf(r0*r0 + r1*r1 + r2*r2 + 1e-12f);
    const float inv = 1.0f / th;
    const float x = r0*inv, y = r1*inv, z = r2*inv;
    const float s = sinf(th), c = cosf(th), t = 1.0f - c;
    float* R = &sR[tid*9];
    R[0] = 1.0f - t*(y*y + z*z);
    R[1] = -s*z + t*x*y;
    R[2] =  s*y + t*x*z;
    R[3] =  s*z + t*x*y;
    R[4] = 1.0f - t*(x*x + z*z);
    R[5] = -s*x + t*y*z;
    R[6] = -s*y + t*x*z;
    R[7] =  s*x + t*y*z;
    R[8] = 1.0f - t*(x*x + y*y);
  }
  __syncthreads();

  // coefficient vector for the channel-collapse pass
  if (tid == 0)  coeff[b*COEFFN + 0] = 1.0f;
  if (tid < 10)  coeff[b*COEFFN + 1 + tid] = sBeta[tid];
  if (tid < 207) {
    const int j = tid/9 + 1, rc = tid%9;
    const float diag = (rc == 0 || rc == 4 || rc == 8) ? 1.0f : 0.0f;
    coeff[b*COEFFN + 11 + tid] = sR[j*9 + rc] - diag;  // pose_map
  }

  // joints: j = Jreg @ (v_template + smpl_shapedirs . beta)
  float acc[NJ*3];
#pragma unroll
  for (int i = 0; i < NJ*3; ++i) acc[i] = 0.0f;
  for (int v = tid; v < NVERT; v += 256) {
    float vd[3];
#pragma unroll
    for (int d = 0; d < 3; ++d) {
      float t = v_template[v*3 + d];
      const float* sd = smpl_shapedirs + (v*3 + d)*10;
#pragma unroll
      for (int k = 0; k < 10; ++k) t = fmaf(sd[k], sBeta[k], t);
      vd[d] = t;
    }
#pragma unroll 4
    for (int jj = 0; jj < NJ; ++jj) {
      const float r = Jreg[jj*NVERT + v];
      acc[jj*3+0] = fmaf(r, vd[0], acc[jj*3+0]);
      acc[jj*3+1] = fmaf(r, vd[1], acc[jj*3+1]);
      acc[jj*3+2] = fmaf(r, vd[2], acc[jj*3+2]);
    }
  }
  // deterministic reduction: wave32 shuffle tree + fixed-order cross-wave sum
  const int wave = tid >> 5, lane = tid & 31;
#pragma unroll 1
  for (int i = 0; i < NJ*3; ++i) {
    float v = acc[i];
#pragma unroll
    for (int off = 16; off > 0; off >>= 1) v += __shfl_xor(v, off, 32);
    if (lane == 0) sPart[wave][i] = v;
  }
  __syncthreads();
  if (tid < NJ*3) {
    float v = 0.0f;
#pragma unroll
    for (int w = 0; w < 8; ++w) v += sPart[w][tid];
    sJ[tid] = v;
  }
  __syncthreads();

  // serial kinematic chain (tiny) on one lane
  if (tid == 0) {
    const int parents[NJ] = {-1,0,0,0,1,2,3,4,5,6,7,8,9,9,9,12,13,14,16,17,18,19,20,21};
    float A[NJ][16];
    for (int i = 0; i < NJ; ++i) {
      float rel[16];
      const int p = parents[i];
      float tx, ty, tz;
      if (i == 0) { tx = sJ[0]; ty = sJ[1]; tz = sJ[2]; }
      else {
        tx = sJ[i*3+0] - sJ[p*3+0];
        ty = sJ[i*3+1] - sJ[p*3+1];
        tz = sJ[i*3+2] - sJ[p*3+2];
      }
#pragma unroll
      for (int r = 0; r < 3; ++r) {
        rel[r*4+0] = sR[i*9 + r*3 + 0];
        rel[r*4+1] = sR[i*9 + r*3 + 1];
        rel[r*4+2] = sR[i*9 + r*3 + 2];
      }
      rel[3] = tx; rel[7] = ty; rel[11] = tz;
      rel[12] = 0.0f; rel[13] = 0.0f; rel[14] = 0.0f; rel[15] = 1.0f;
      if (i == 0) { for (int u = 0; u < 16; ++u) A[0][u] = rel[u]; }
      else mat4mul(A[i], A[p], rel);
    }
    for (int jj = 0; jj < NJ; ++jj) {
      const float jx = sJ[jj*3+0], jy = sJ[jj*3+1], jz = sJ[jj*3+2];
#pragma unroll
      for (int r = 0; r < 4; ++r) {
        const float t = A[jj][r*4+0]*jx + A[jj][r*4+1]*jy + A[jj][r*4+2]*jz;
        A[jj][r*4+3] -= t;
      }
    }
    for (int k = 0; k < 16; ++k)
      for (int jj = 0; jj < NJ; ++jj)
        A16w[b*16*NJ + k*NJ + jj] = A[jj][k];
  }
}

// ---------------------------------------------------------------------------
// Kernel 2: collapse 654 grid channels -> 3 channels per batch (streaming,
// b128 coalesced; single pass over the 711 MB grid serves both batches).
// Gc[b][d][voxel] = closest[d] + sum_k shapedirs[d*10+k]*beta[k]
//                              + sum_k posedirs[d*207+k]*pose_map[k]
// ---------------------------------------------------------------------------
__global__ __launch_bounds__(256)
void collapse_kernel(const float* __restrict__ closest,
                     const float* __restrict__ shaped,
                     const float* __restrict__ posed,
                     const float* __restrict__ coeff,
                     float* __restrict__ Gc, int Bn)
{
  __shared__ float sc[2*COEFFN];
  for (int i = threadIdx.x; i < 2*COEFFN; i += 256) sc[i] = 0.0f;
  __syncthreads();
  for (int i = threadIdx.x; i < Bn*COEFFN; i += 256) sc[i] = coeff[i];
  __syncthreads();

  const F4* c4 = (const F4*)closest;
  const F4* s4 = (const F4*)shaped;
  const F4* p4 = (const F4*)posed;
  F4* g4 = (F4*)Gc;

  const int q = blockIdx.x*256 + threadIdx.x;
  if (q >= NVOX4) return;

#pragma unroll 1
  for (int d = 0; d < 3; ++d) {
    F4 a0, a1, g;
    g = c4[d*NVOX4 + q];
#pragma unroll
    for (int u = 0; u < 4; ++u) { a0.v[u] = g.v[u]*sc[0]; a1.v[u] = g.v[u]*sc[COEFFN]; }
#pragma unroll
    for (int k = 0; k < 10; ++k) {
      g = s4[(d*10 + k)*NVOX4 + q];
      const float c0 = sc[1+k], c1 = sc[COEFFN+1+k];
#pragma unroll
      for (int u = 0; u < 4; ++u) {
        a0.v[u] = fmaf(g.v[u], c0, a0.v[u]);
        a1.v[u] = fmaf(g.v[u], c1, a1.v[u]);
      }
    }
#pragma unroll 3
    for (int k = 0; k < 207; ++k) {
      if ((k & 15) == 0 && k + 16 < 207)
        __builtin_prefetch((const void*)(p4 + (size_t)(d*207 + k + 16)*NVOX4 + q), 0, 1);
      g = p4[(size_t)(d*207 + k)*NVOX4 + q];
      const float c0 = sc[11+k], c1 = sc[COEFFN+11+k];
#pragma unroll
      for (int u = 0; u < 4; ++u) {
        a0.v[u] = fmaf(g.v[u], c0, a0.v[u]);
        a1.v[u] = fmaf(g.v[u], c1, a1.v[u]);
      }
    }
    g4[d*NVOX4 + q] = a0;
    if (Bn > 1) g4[(3 + d)*NVOX4 + q] = a1;
  }
}

// ---------------------------------------------------------------------------
// Kernel 3: per-point trilinear gather (3 collapsed ch + 24 skin ch) then the
// skinning transform T = A16 . w as a WMMA GEMM (M=16 pts, N=16 comps, K=24)
// using v_wmma_f32_16x16x4_f32 (f32 path keeps reference precision).
// ---------------------------------------------------------------------------
__global__ __launch_bounds__(256)
void points_kernel(const float* __restrict__ points, const float* __restrict__ trans,
                   const float* __restrict__ scale, const float* __restrict__ center,
                   const float* __restrict__ Gc, const float* __restrict__ skin,
                   const float* __restrict__ A16w, float* __restrict__ out, int Np)
{
  __shared__ float sA16[16*NJ];     // [component 16][joint 24]
  __shared__ float sw[256*25];      // skin weights, padded stride
  __shared__ float sh[256*4];       // homogeneous posed points
  __shared__ float sT[256*16];      // per-point 4x4 transform

  const int tid = threadIdx.x;
  const int g   = blockIdx.x*256 + tid;      // block never straddles a batch
  const int b   = (blockIdx.x*256) / Np;

  for (int i = tid; i < 16*NJ; i += 256) sA16[i] = A16w[b*16*NJ + i];

  const float s = scale[0];
  const float px = fmaf(points[g*3+0], s, center[0]);
  const float py = fmaf(points[g*3+1], s, center[1]);
  const float pz = fmaf(points[g*3+2], s, center[2]);

  const float cx = ((px + 1.0f)*RES - 1.0f)*0.5f;
  const float cy = ((py + 1.0f)*RES - 1.0f)*0.5f;
  const float cz = ((pz + 1.0f)*RES - 1.0f)*0.5f;
  const float fx = floorf(cx), fy = floorf(cy), fz = floorf(cz);
  const float w1x = cx - fx, w0x = 1.0f - w1x;
  const float w1y = cy - fy, w0y = 1.0f - w1y;
  const float w1z = cz - fz, w0z = 1.0f - w1z;
  const int ix = (int)fx, iy = (int)fy, iz = (int)fz;

  int   offs[8];
  float wts[8];
#pragma unroll
  for (int cc = 0; cc < 8; ++cc) {
    const int dx = cc >> 2, dy = (cc >> 1) & 1, dz = cc & 1;
    const int jx = ix + dx, jy = iy + dy, jz = iz + dz;
    const bool ok = (jx >= 0) & (jx < RES) & (jy >= 0) & (jy < RES) & (jz >= 0) & (jz < RES);
    const float w = (dx ? w1x : w0x)*(dy ? w1y : w0y)*(dz ? w1z : w0z);
    wts[cc] = ok ? w : 0.0f;
    const int cjx = jx < 0 ? 0 : (jx > RES-1 ? RES-1 : jx);
    const int cjy = jy < 0 ? 0 : (jy > RES-1 ? RES-1 : jy);
    const int cjz = jz < 0 ? 0 : (jz > RES-1 ? RES-1 : jz);
    offs[cc] = (cjx*RES + cjy)*RES + cjz;
  }

  const float* gb = Gc + (size_t)b*3*NVOX;
  float p0 = 0.0f, p1 = 0.0f, p2 = 0.0f;
#pragma unroll
  for (int cc = 0; cc < 8; ++cc) {
    const float w = wts[cc]; const int o = offs[cc];
    p0 = fmaf(w, gb[o],        p0);
    p1 = fmaf(w, gb[NVOX+o],   p1);
    p2 = fmaf(w, gb[2*NVOX+o], p2);
  }
  float wj[NJ];
#pragma unroll
  for (int j = 0; j < NJ; ++j) {
    float a = 0.0f;
#pragma unroll
    for (int cc = 0; cc < 8; ++cc) a = fmaf(wts[cc], skin[j*NVOX + offs[cc]], a);
    wj[j] = a;
    sw[tid*25 + j] = a;
  }
  sh[tid*4+0] = p0; sh[tid*4+1] = p1; sh[tid*4+2] = p2; sh[tid*4+3] = 1.0f;
  __syncthreads();

  float Treg[16];
#if defined(__gfx1250__) && __has_builtin(__builtin_amdgcn_wmma_f32_16x16x4_f32)
  {
    // D[m][n] = sum_j W[m][j] * A16[n][j] ; A=W (16x24), B=A16^T (24x16)
    const int wave = tid >> 5, lane = tid & 31;
    const int n  = lane & 15;   // output component / B column
    const int kh = lane >> 4;   // K-half per ISA f32 A/B layout
#pragma unroll
    for (int t = 0; t < 2; ++t) {
      const int rowBase = wave*32 + t*16;
      const int m = rowBase + n;            // A-matrix row for this lane
      v8f acc = {};
#pragma unroll
      for (int k0 = 0; k0 < NJ; k0 += 4) {
        v2f av; av.x = sw[m*25 + k0 + 2*kh];      av.y = sw[m*25 + k0 + 1 + 2*kh];
        v2f bv; bv.x = sA16[n*NJ + k0 + 2*kh];    bv.y = sA16[n*NJ + k0 + 1 + 2*kh];
        acc = __builtin_amdgcn_wmma_f32_16x16x4_f32(false, av, false, bv,
                                                    (short)0, acc, false, false);
      }
      // D layout: vgpr i holds M = kh*8 + i, N = lane%16
#pragma unroll
      for (int i = 0; i < 8; ++i) sT[(rowBase + kh*8 + i)*16 + n] = acc[i];
    }
    __syncthreads();
#pragma unroll
    for (int k = 0; k < 16; ++k) Treg[k] = sT[tid*16 + k];
  }
#else
  // scalar fallback (still correct, no WMMA emitted)
#pragma unroll
  for (int k = 0; k < 16; ++k) {
    float t = 0.0f;
#pragma unroll
    for (int j = 0; j < NJ; ++j) t = fmaf(sA16[k*NJ + j], wj[j], t);
    Treg[k] = t;
  }
#endif

  const float h0 = sh[tid*4+0], h1 = sh[tid*4+1], h2 = sh[tid*4+2];
  const float o0 = Treg[0]*h0 + Treg[1]*h1 + Treg[2]*h2  + Treg[3];
  const float o1 = Treg[4]*h0 + Treg[5]*h1 + Treg[6]*h2  + Treg[7];
  const float o2 = Treg[8]*h0 + Treg[9]*h1 + Treg[10]*h2 + Treg[11];
  out[g*3+0] = o0 + trans[b*3+0];
  out[g*3+1] = o1 + trans[b*3+1];
  out[g*3+2] = o2 + trans[b*3+2];
}

// ---------------------------------------------------------------------------
extern "C" void kernel_launch(void* const* d_in, const int* in_sizes, int n_in,
                              void* d_out, int out_size, void* d_ws, size_t ws_size,
                              hipStream_t stream)
{
  const float* points  = (const float*)d_in[0];
  const float* pose    = (const float*)d_in[1];
  const float* beta    = (const float*)d_in[2];
  const float* trans   = (const float*)d_in[3];
  const float* scale   = (const float*)d_in[4];
  const float* center  = (const float*)d_in[5];
  const float* closest = (const float*)d_in[6];
  const float* shaped  = (const float*)d_in[7];
  const float* posed   = (const float*)d_in[8];
  const float* skin    = (const float*)d_in[9];
  const float* vtem    = (const float*)d_in[10];
  const float* sdirs   = (const float*)d_in[11];
  const float* jreg    = (const float*)d_in[12];
  float* out = (float*)d_out;

  const int Bn = in_sizes[1] / 72;          // batch (2)
  const int Np = in_sizes[0] / (3 * Bn);    // points per batch (32768)

  float* ws    = (float*)d_ws;
  float* coeff = ws;                                         // Bn*218
  float* A16w  = coeff + (size_t)Bn * COEFFN;                // Bn*384
  size_t gcOff = ((size_t)Bn * (COEFFN + 16*NJ) + 3) & ~(size_t)3;  // 16B-align
  float* Gc    = ws + gcOff;                                 // Bn*3*NVOX

  prep_kernel<<<Bn, 256, 0, stream>>>(pose, beta, vtem, sdirs, jreg, coeff, A16w);
  collapse_kernel<<<NVOX4/256, 256, 0, stream>>>(closest, shaped, posed, coeff, Gc, Bn);
  const int P = Bn * Np;
  points_kernel<<<P/256, 256, 0, stream>>>(points, trans, scale, center,
                                           Gc, skin, A16w, out, Np);
}